// GCNLayer_30116310680316
// MI455X (gfx1250) — compile-verified
//
#include <hip/hip_runtime.h>
#include <hip/hip_bf16.h>

typedef __attribute__((ext_vector_type(2))) float v2f;
typedef __attribute__((ext_vector_type(8))) float v8f;

#define D_FEAT 48
#define EPW 128  // edges per wave

__global__ __launch_bounds__(256) void spmm_zero_kernel(float* __restrict__ out, int n) {
    int i = blockIdx.x * blockDim.x + threadIdx.x;
    if (i < n) out[i] = 0.0f;
}

// Fallback: one edge, all 32 lanes cooperate across the 48 features.
__device__ __forceinline__ void scalar_edge(const int* __restrict__ rows,
                                            const int* __restrict__ cols,
                                            const float* __restrict__ vals,
                                            const float* __restrict__ x,
                                            float* __restrict__ out,
                                            int idx, int lane) {
    int r = rows[idx];
    int c = cols[idx];
    float v = vals[idx];
    atomicAdd(&out[r * D_FEAT + lane], v * x[c * D_FEAT + lane]);
    if (lane < 16)
        atomicAdd(&out[r * D_FEAT + 32 + lane], v * x[c * D_FEAT + 32 + lane]);
}

__global__ __launch_bounds__(256) void spmm_wmma_kernel(const int* __restrict__ rows,
                                                        const int* __restrict__ cols,
                                                        const float* __restrict__ vals,
                                                        const float* __restrict__ x,
                                                        float* __restrict__ out,
                                                        int nEdges) {
    const int lane = threadIdx.x & 31;
    const int half = lane >> 4;   // which 16-lane half
    const int l16  = lane & 15;

    const int waveId = (blockIdx.x * blockDim.x + threadIdx.x) >> 5;
    int e0 = waveId * EPW;
    if (e0 >= nEdges) return;
    int eEnd = min(e0 + EPW, nEdges);

    const v8f vzero = {};
    v8f acc0 = {}, acc1 = {}, acc2 = {};   // 16 rows x 48 feats (3 tiles of 16)

    int win_lo = rows[e0];   // window covers rows [win_lo, win_lo+15]
    int hi_row = win_lo - 1; // max row actually accumulated (empty => < win_lo)

    // Flush the 16x48 accumulator tile into out[] with f32 atomics,
    // touching only rows that were actually accumulated.
    auto flush = [&]() {
#pragma unroll
        for (int vv = 0; vv < 8; ++vv) {
            int row = win_lo + vv + 8 * half;  // C/D layout: VGPR v, halves hold M=v / M=v+8
            if (row <= hi_row) {
                float* o = out + row * D_FEAT + l16;
                atomicAdd(o + 0,  acc0[vv]);
                atomicAdd(o + 16, acc1[vv]);
                atomicAdd(o + 32, acc2[vv]);
            }
        }
        acc0 = vzero; acc1 = vzero; acc2 = vzero;
    };

    int e = e0;
    for (; e + 4 <= eEnd; e += 4) {
        // Prefetch the edge stream ahead (global_prefetch_b8).
        __builtin_prefetch(rows + e + 256, 0, 0);
        __builtin_prefetch(cols + e + 256, 0, 0);
        __builtin_prefetch(vals + e + 256, 0, 0);

        // Wave-uniform window management (rows are sorted ascending).
        int rlo = rows[e];
        int rhi = rows[e + 3];
        if (rhi - win_lo > 15) {
            flush();
            win_lo = rlo;
            hi_row = win_lo - 1;
        }
        if (rhi - win_lo > 15) {
            // 4 edges span >16 rows: rare pathological case, do it directly.
#pragma unroll
            for (int k = 0; k < 4; ++k)
                scalar_edge(rows, cols, vals, x, out, e + k, lane);
            continue;
        }

        // Each lane owns edges k = 2*half and 2*half+1 for both A and B operands.
        int ek = e + 2 * half;
        int r0 = rows[ek],     r1 = rows[ek + 1];
        int c0 = cols[ek],     c1 = cols[ek + 1];
        float v0 = vals[ek],   v1 = vals[ek + 1];

        // A (16x4 f32): A[i][k] = (local row of edge k == i) ? val[k] : 0
        v2f a;
        a.x = (r0 - win_lo == l16) ? v0 : 0.0f;
        a.y = (r1 - win_lo == l16) ? v1 : 0.0f;

        // B (4x16 f32) per feature tile: B[k][n] = x[cols[k]][f + n]
        const float* xr0 = x + c0 * D_FEAT + l16;
        const float* xr1 = x + c1 * D_FEAT + l16;
        v2f b0, b1, b2;
        b0.x = xr0[0];   b0.y = xr1[0];
        b1.x = xr0[16];  b1.y = xr1[16];
        b2.x = xr0[32];  b2.y = xr1[32];

        // D(16x16) += A(16x4) x B(4x16): segment-sum of 4 edges into the row window.
        acc0 = __builtin_amdgcn_wmma_f32_16x16x4_f32(false, a, false, b0, (short)0, acc0, false, false);
        acc1 = __builtin_amdgcn_wmma_f32_16x16x4_f32(false, a, false, b1, (short)0, acc1, false, false);
        acc2 = __builtin_amdgcn_wmma_f32_16x16x4_f32(false, a, false, b2, (short)0, acc2, false, false);

        hi_row = rhi;
    }

    flush();

    // Tail edges (chunk size not a multiple of 4).
    for (; e < eEnd; ++e)
        scalar_edge(rows, cols, vals, x, out, e, lane);
}

extern "C" void kernel_launch(void* const* d_in, const int* in_sizes, int n_in,
                              void* d_out, int out_size, void* d_ws, size_t ws_size,
                              hipStream_t stream) {
    const int*   rows = (const int*)  d_in[0];
    const int*   cols = (const int*)  d_in[1];
    const float* vals = (const float*)d_in[2];
    const float* x    = (const float*)d_in[3];
    float*       out  = (float*)d_out;

    const int nEdges = in_sizes[0];

    // Zero the (poisoned) output first; we accumulate with atomics.
    int zb = (out_size + 255) / 256;
    spmm_zero_kernel<<<zb, 256, 0, stream>>>(out, out_size);

    int waves   = (nEdges + EPW - 1) / EPW;
    int threads = 256;                       // 8 waves per block, each independent
    int blocks  = (waves * 32 + threads - 1) / threads;
    spmm_wmma_kernel<<<blocks, threads, 0, stream>>>(rows, cols, vals, x, out, nEdges);
}